// CapsuleNetwork_55362128445565
// MI455X (gfx1250) — compile-verified
//
#include <hip/hip_runtime.h>
#include <stdint.h>

// Problem constants (match reference)
#define BB 2048
#define SS 200
#define DD 64
#define KK 4
#define ND 256          // K*D
#define HIS_STRIDE 68   // padded row stride (dwords) for his tile: 4-bank row step
#define HAT_STRIDE 260  // padded row stride (dwords) for hat tile
#define MTILES 13       // ceil(200/16)

typedef float v2f __attribute__((ext_vector_type(2)));
typedef float v8f __attribute__((ext_vector_type(8)));

__launch_bounds__(256, 1)
__global__ void capsule_mind_kernel(const float* __restrict__ his,
                                    const float* __restrict__ eb,
                                    const int*   __restrict__ mask,
                                    const float* __restrict__ Wg,
                                    const float* __restrict__ bg,
                                    float* __restrict__ outI,   // [B,K,D]
                                    float* __restrict__ outR) { // [B,D]
  __shared__ __attribute__((aligned(16))) float sHis[208 * HIS_STRIDE]; // 56.6 KB
  __shared__ __attribute__((aligned(16))) float sHat[SS * HAT_STRIDE]; // 208 KB
  __shared__ float sCw[KK * SS];
  __shared__ float sSw[KK * SS];
  __shared__ float sV[KK * DD];
  __shared__ float sInt[KK * DD];
  __shared__ int   sMask[SS];
  __shared__ float sLog[KK];
  __shared__ int   sIdx;

  const int tid = threadIdx.x;
  const int b   = blockIdx.x;

  // ---------------- Stage his[b] (200x64 f32) -> LDS via async copies --------
  const float* hisB = his + (size_t)b * (SS * DD);
  for (int c = tid; c < SS * (DD / 4); c += 256) {   // 3200 x 16B chunks
    const int row = c >> 4;                          // 16 chunks per row
    const int seg = (c & 15) << 2;                   // dword column
    unsigned ldsOff = (unsigned)(uintptr_t)(&sHis[row * HIS_STRIDE + seg]);
    unsigned long long ga = (unsigned long long)(uintptr_t)(hisB + row * DD + seg);
    asm volatile("global_load_async_to_lds_b128 %0, %1, off"
                 :: "v"(ldsOff), "v"(ga) : "memory");
  }
  // While copies fly: zero pad rows (M-tile 13 covers rows 200..207), init cw, mask
  for (int i = tid; i < 8 * HIS_STRIDE; i += 256) sHis[SS * HIS_STRIDE + i] = 0.0f;
  for (int s = tid; s < KK * SS; s += 256) sCw[s] = 0.0f;
  for (int s = tid; s < SS; s += 256) sMask[s] = mask[b * SS + s];
  asm volatile("s_wait_asynccnt 0" ::: "memory");
  __syncthreads();

  // ---------------- GEMM: hat = his @ W + bias via V_WMMA_F32_16X16X4_F32 ----
  const int lane = tid & 31;
  const int wv   = tid >> 5;      // 8 waves
  const int l16  = lane & 15;
  const int half = lane >> 4;     // 0 | 1

  for (int nt = wv; nt < 16; nt += 8) {
    const int ncol = nt * 16 + l16;
    v2f bf[16];
#pragma unroll
    for (int kk = 0; kk < 16; ++kk) {            // B frag: rows k+2h, k+2h+1
      const int kr = kk * 4 + 2 * half;
      bf[kk][0] = Wg[kr * ND + ncol];
      bf[kk][1] = Wg[(kr + 1) * ND + ncol];
    }
    const float bias = bg[ncol];
    for (int mt = 0; mt < MTILES; ++mt) {
      const int m0 = mt * 16;
      v8f acc = {0.f, 0.f, 0.f, 0.f, 0.f, 0.f, 0.f, 0.f};
#pragma unroll
      for (int kk = 0; kk < 16; ++kk) {
        // A frag: his[m0+l16][k+2h : k+2h+1] -- 8B-aligned ds_load_b64
        v2f a = *(const v2f*)(&sHis[(m0 + l16) * HIS_STRIDE + kk * 4 + 2 * half]);
        acc = __builtin_amdgcn_wmma_f32_16x16x4_f32(
            false, a, false, bf[kk], (short)0, acc, false, false);
      }
#pragma unroll
      for (int r = 0; r < 8; ++r) {              // C row = m0 + r + 8*half
        const int row = m0 + r + 8 * half;
        if (row < SS) sHat[row * HAT_STRIDE + ncol] = acc[r] + bias;
      }
    }
  }
  __syncthreads();

  // ---------------- Dynamic routing (3 iterations), all from LDS -------------
  const int k = tid >> 6;         // interest index 0..3 (wave-uniform)
  const int d = tid & 63;

  for (int it = 0; it < 3; ++it) {
    // softmax over K per position, then mask-zero
    for (int s = tid; s < SS; s += 256) {
      float c0 = sCw[s], c1 = sCw[SS + s], c2 = sCw[2 * SS + s], c3 = sCw[3 * SS + s];
      float mx = fmaxf(fmaxf(c0, c1), fmaxf(c2, c3));
      float e0 = __expf(c0 - mx), e1 = __expf(c1 - mx);
      float e2 = __expf(c2 - mx), e3 = __expf(c3 - mx);
      float inv = 1.0f / (e0 + e1 + e2 + e3);
      if (sMask[s] == 0) inv = 0.0f;
      sSw[s] = e0 * inv; sSw[SS + s] = e1 * inv;
      sSw[2 * SS + s] = e2 * inv; sSw[3 * SS + s] = e3 * inv;
    }
    __syncthreads();

    // interest[k][d] = sum_s sw[k][s] * hat[s][k*64+d]
    float acc = 0.0f;
    {
      const float* swk = &sSw[k * SS];
      const float* hk  = &sHat[k * DD + d];
#pragma unroll 4
      for (int s = 0; s < SS; ++s) acc += swk[s] * hk[s * HAT_STRIDE];
    }
    sV[tid] = acc;
    __syncthreads();

    // squash
    float n2 = 0.0f;
#pragma unroll 8
    for (int d2 = 0; d2 < DD; ++d2) { float v = sV[k * DD + d2]; n2 += v * v; }
    const float inter = acc * (n2 / (1.0f + n2) / sqrtf(n2 + 1e-9f));
    sInt[tid] = inter;
    if (it == 2) outI[(size_t)b * ND + tid] = inter;
    __syncthreads();

    if (it < 2) {
      // cw[k][s] += sum_d hat[s][k*64+d] * interest[k][d]
      for (int p = tid; p < KK * SS; p += 256) {
        const int k2 = p / SS;
        const int s2 = p - k2 * SS;
        const float* hp = &sHat[s2 * HAT_STRIDE + k2 * DD];
        const float* ip = &sInt[k2 * DD];
        float a2 = 0.0f;
#pragma unroll 8
        for (int d2 = 0; d2 < DD; ++d2) a2 += hp[d2] * ip[d2];
        sCw[p] += a2;
      }
      __syncthreads();
    }
  }

  // ---------------- Readout: argmax over interest-vs-target logits -----------
  if (tid < KK) {
    const float* e = eb + (size_t)b * DD;
    float a = 0.0f;
#pragma unroll 8
    for (int d2 = 0; d2 < DD; ++d2) a += sInt[tid * DD + d2] * e[d2];
    sLog[tid] = a;
  }
  __syncthreads();
  if (tid == 0) {
    int best = 0; float bv = sLog[0];
#pragma unroll
    for (int k2 = 1; k2 < KK; ++k2) if (sLog[k2] > bv) { bv = sLog[k2]; best = k2; }
    sIdx = best;
  }
  __syncthreads();
  if (tid < DD) outR[(size_t)b * DD + tid] = sInt[sIdx * DD + tid];
}

extern "C" void kernel_launch(void* const* d_in, const int* in_sizes, int n_in,
                              void* d_out, int out_size, void* d_ws, size_t ws_size,
                              hipStream_t stream) {
  const float* his  = (const float*)d_in[0];  // [B,S,D]
  const float* eb   = (const float*)d_in[1];  // [B,D]
  const int*   mask = (const int*)d_in[2];    // [B,S]
  const float* W    = (const float*)d_in[3];  // [D,K*D]
  const float* bias = (const float*)d_in[4];  // [K*D]
  float* out  = (float*)d_out;
  float* outI = out;                 // [B,K,D] = 524288 floats
  float* outR = out + (size_t)BB * KK * DD;  // [B,D]
  capsule_mind_kernel<<<dim3(BB), dim3(256), 0, stream>>>(his, eb, mask, W, bias, outI, outR);
}